// NNGrouper_65000035057786
// MI455X (gfx1250) — compile-verified
//
#include <hip/hip_runtime.h>
#include <float.h>
#include <math.h>

typedef __attribute__((ext_vector_type(2))) float v2f;
typedef __attribute__((ext_vector_type(8))) float v8f;

#define B_   16
#define N_   16384
#define F_   64
#define K_   512
#define COUT 68   // 3 (normalized offset) + 1 (dist) + 64 (features)

// Probe the async memory->LDS DMA path (GLOBAL_LOAD_ASYNC_TO_LDS_B32, ASYNCcnt).
#if defined(__has_builtin)
#  if __has_builtin(__builtin_amdgcn_global_load_async_to_lds_b32) && \
      __has_builtin(__builtin_amdgcn_s_wait_asynccnt)
#    define USE_ASYNC_LDS 1
#  endif
#endif
#ifndef USE_ASYNC_LDS
#  define USE_ASYNC_LDS 0
#endif

#if USE_ASYNC_LDS
typedef __attribute__((address_space(1))) int gas_int;  // global (prints as __device__ int*)
typedef __attribute__((address_space(3))) int las_int;  // LDS    (prints as __shared__ int*)
#endif

// ---------------------------------------------------------------------------
// Kernel 1: deterministic farthest point sampling (first index = 0).
// One block per batch, 1024 threads, each thread owns 16 contiguous points
// and their running min-distance entirely in VGPRs. The argmax reduction
// carries the winning point's coordinates, so the 512-step serial loop never
// touches global memory (only tiny LDS broadcasts).
// ---------------------------------------------------------------------------
__global__ __launch_bounds__(1024) void fps_kernel(const float* __restrict__ xyz,
                                                   float* __restrict__ centers /*[B,K,3]*/) {
  __shared__ float s_val[32];
  __shared__ int   s_idx[32];
  __shared__ float s_x[32], s_y[32], s_z[32];
  __shared__ float s_cur[3];

  const int b    = blockIdx.x;
  const int tid  = threadIdx.x;
  const int lane = tid & 31;
  const int wv   = tid >> 5;

  const float* base = xyz + (size_t)b * N_ * 3;

  float px[16], py[16], pz[16], md[16];
#pragma unroll
  for (int i = 0; i < 16; ++i) {
    const int n = tid * 16 + i;
    px[i] = base[n * 3 + 0];
    py[i] = base[n * 3 + 1];
    pz[i] = base[n * 3 + 2];
    md[i] = FLT_MAX;
  }

  if (tid == 0) { s_cur[0] = px[0]; s_cur[1] = py[0]; s_cur[2] = pz[0]; }
  __syncthreads();

  for (int k = 0; k < K_; ++k) {
    const float cx = s_cur[0], cy = s_cur[1], cz = s_cur[2];
    if (tid == 0) {
      centers[((size_t)b * K_ + k) * 3 + 0] = cx;
      centers[((size_t)b * K_ + k) * 3 + 1] = cy;
      centers[((size_t)b * K_ + k) * 3 + 2] = cz;
    }

    float bv = -1.0f; int bn = 0;
    float bx = 0.f, by = 0.f, bz = 0.f;
#pragma unroll
    for (int i = 0; i < 16; ++i) {
      const float dx = px[i] - cx, dy = py[i] - cy, dz = pz[i] - cz;
      float d = fmaf(dx, dx, fmaf(dy, dy, dz * dz));
      d = fminf(d, md[i]);
      md[i] = d;
      const bool upd = d > bv;           // strict '>' keeps lowest index on tie
      bv = upd ? d : bv;
      bn = upd ? tid * 16 + i : bn;
      bx = upd ? px[i] : bx;
      by = upd ? py[i] : by;
      bz = upd ? pz[i] : bz;
    }
    // wave-level argmax (carrying coordinates), lowest index wins ties
#pragma unroll
    for (int off = 16; off >= 1; off >>= 1) {
      const float ov = __shfl_xor(bv, off, 32);
      const int   on = __shfl_xor(bn, off, 32);
      const float ox = __shfl_xor(bx, off, 32);
      const float oy = __shfl_xor(by, off, 32);
      const float oz = __shfl_xor(bz, off, 32);
      const bool take = (ov > bv) || (ov == bv && on < bn);
      bv = take ? ov : bv; bn = take ? on : bn;
      bx = take ? ox : bx; by = take ? oy : by; bz = take ? oz : bz;
    }
    if (lane == 0) { s_val[wv] = bv; s_idx[wv] = bn; s_x[wv] = bx; s_y[wv] = by; s_z[wv] = bz; }
    __syncthreads();
    if (wv == 0) {
      bv = s_val[lane]; bn = s_idx[lane];
      bx = s_x[lane];  by = s_y[lane];  bz = s_z[lane];
#pragma unroll
      for (int off = 16; off >= 1; off >>= 1) {
        const float ov = __shfl_xor(bv, off, 32);
        const int   on = __shfl_xor(bn, off, 32);
        const float ox = __shfl_xor(bx, off, 32);
        const float oy = __shfl_xor(by, off, 32);
        const float oz = __shfl_xor(bz, off, 32);
        const bool take = (ov > bv) || (ov == bv && on < bn);
        bv = take ? ov : bv; bn = take ? on : bn;
        bx = take ? ox : bx; by = take ? oy : by; bz = take ? oz : bz;
      }
      if (lane == 0) { s_cur[0] = bx; s_cur[1] = by; s_cur[2] = bz; }
    }
    __syncthreads();
  }
}

// ---------------------------------------------------------------------------
// Kernel 2: nearest-center via V_WMMA_F32_16X16X4_F32.
// Score(p, c) = |c|^2 - 2 p.c  (same argmin as squared distance), computed as
// a rank-4 GEMM: A row = (-2x, -2y, -2z, 1), B col = (cx, cy, cz, |c|^2).
// Each wave owns 16 points x all 512 centers (32 WMMA ops), then reduces the
// per-lane argmin across each 16-lane half. Block = 8 waves = 128 points.
// Centers are streamed to LDS with GLOBAL_LOAD_ASYNC_TO_LDS when available.
// ---------------------------------------------------------------------------
__global__ __launch_bounds__(256) void knn_group_kernel(
    const float* __restrict__ xyz, const float* __restrict__ features,
    const float* __restrict__ centers, float* __restrict__ out_gf,
    float* __restrict__ out_nn) {
  __shared__ float scx[K_], scy[K_], scz[K_], sc2[K_];
  __shared__ float pdata[128][4];
  __shared__ int   s_nn[128];
#if USE_ASYNC_LDS
  __shared__ float stage[K_ * 3];
#endif

  const int tid    = threadIdx.x;
  const int lane   = tid & 31;
  const int wv     = tid >> 5;
  const int b      = blockIdx.x >> 7;   // 128 chunks per batch
  const int chunk  = blockIdx.x & 127;
  const int n_base = chunk * 128;

  const float* cb = centers + (size_t)b * K_ * 3;

#if USE_ASYNC_LDS
  // Async DMA the raw [K,3] centers into LDS staging (ASYNCcnt-tracked),
  // then build the SoA + |c|^2 table from LDS.
  for (int i = tid; i < K_ * 3; i += 256) {
    __builtin_amdgcn_global_load_async_to_lds_b32(
        (gas_int*)(cb + i),
        (las_int*)(&stage[i]),
        0, 0);
  }
  __builtin_amdgcn_s_wait_asynccnt(0);
  __syncthreads();
  for (int k = tid; k < K_; k += 256) {
    const float x = stage[k * 3 + 0], y = stage[k * 3 + 1], z = stage[k * 3 + 2];
    scx[k] = x; scy[k] = y; scz[k] = z;
    sc2[k] = fmaf(x, x, fmaf(y, y, z * z));
  }
#else
  for (int k = tid; k < K_; k += 256) {
    const float x = cb[k * 3 + 0], y = cb[k * 3 + 1], z = cb[k * 3 + 2];
    scx[k] = x; scy[k] = y; scz[k] = z;
    sc2[k] = fmaf(x, x, fmaf(y, y, z * z));
  }
#endif
  __syncthreads();

  // A-matrix (16x4 f32, 2 VGPRs): lanes 0-15 hold K=0,1; lanes 16-31 hold K=2,3
  const int col = lane & 15;
  const int m   = n_base + wv * 16 + col;
  const float* pp = xyz + ((size_t)b * N_ + m) * 3;
  v2f a;
  if (lane < 16) { a.x = -2.0f * pp[0]; a.y = -2.0f * pp[1]; }
  else           { a.x = -2.0f * pp[2]; a.y = 1.0f; }

  float bd[8]; int bi[8];
#pragma unroll
  for (int j = 0; j < 8; ++j) { bd[j] = FLT_MAX; bi[j] = 0; }

  // Partial unroll: enough WMMAs in flight to pipeline, short mask live-ranges
  // (full unroll caused SGPR spills to VGPR lanes).
#pragma unroll 4
  for (int nt = 0; nt < 32; ++nt) {
    const int n = nt * 16 + col;
    v2f bvec;
    if (lane < 16) { bvec.x = scx[n]; bvec.y = scy[n]; }   // rows K=0,1
    else           { bvec.x = scz[n]; bvec.y = sc2[n]; }   // rows K=2,3
    v8f c = {};
    c = __builtin_amdgcn_wmma_f32_16x16x4_f32(
        /*neg_a=*/false, a, /*neg_b=*/false, bvec,
        /*c_mod=*/(short)0, c, /*reuse_a=*/false, /*reuse_b=*/false);
#pragma unroll
    for (int j = 0; j < 8; ++j) {
      const float s = c[j];
      const bool upd = s < bd[j];        // strict '<' keeps lowest index on tie
      bd[j] = upd ? s : bd[j];
      bi[j] = upd ? n : bi[j];
    }
  }
  // reduce argmin across the 16 lanes of each half (xor stays within halves)
#pragma unroll
  for (int off = 1; off <= 8; off <<= 1) {
#pragma unroll
    for (int j = 0; j < 8; ++j) {
      const float od = __shfl_xor(bd[j], off, 32);
      const int   oi = __shfl_xor(bi[j], off, 32);
      const bool take = (od < bd[j]) || (od == bd[j] && oi < bi[j]);
      bd[j] = take ? od : bd[j];
      bi[j] = take ? oi : bi[j];
    }
  }
  // lanes 0-15 hold rows M=0..7 (VGPR j -> M=j); lanes 16-31 hold M=8..15
  if (lane == 0) {
#pragma unroll
    for (int j = 0; j < 8; ++j) s_nn[wv * 16 + j] = bi[j];
  } else if (lane == 16) {
#pragma unroll
    for (int j = 0; j < 8; ++j) s_nn[wv * 16 + 8 + j] = bi[j];
  }
  __syncthreads();

  // per-point offset / distance / normalization
  if (tid < 128) {
    const int n   = n_base + tid;
    const int idx = s_nn[tid];
    const float* q = xyz + ((size_t)b * N_ + n) * 3;
    const float dx = q[0] - scx[idx];
    const float dy = q[1] - scy[idx];
    const float dz = q[2] - scz[idx];
    const float dist = sqrtf(fmaf(dx, dx, fmaf(dy, dy, dz * dz)));
    const float inv  = 1.0f / fmaxf(dist, 1e-8f);
    pdata[tid][0] = dx * inv;
    pdata[tid][1] = dy * inv;
    pdata[tid][2] = dz * inv;
    pdata[tid][3] = dist;
    out_nn[(size_t)b * N_ + n] = (float)idx;
  }
  __syncthreads();

  // coalesced concat write: [nbr_xyz(3), dist(1), features(64)] per point
  for (int t = tid; t < 128 * COUT; t += 256) {
    const int p = t / COUT;
    const int c = t - p * COUT;
    const int n = n_base + p;
    const float v = (c < 4) ? pdata[p][c]
                            : features[((size_t)b * N_ + n) * F_ + (c - 4)];
    out_gf[((size_t)b * N_ + n) * COUT + c] = v;
  }
}

extern "C" void kernel_launch(void* const* d_in, const int* in_sizes, int n_in,
                              void* d_out, int out_size, void* d_ws, size_t ws_size,
                              hipStream_t stream) {
  const float* xyz      = (const float*)d_in[0];
  const float* features = (const float*)d_in[1];
  // d_in[2] = num_groups scalar (fixed to 512 for this problem size)

  float* out         = (float*)d_out;
  float* out_gf      = out;                                   // [B,N,68]
  float* out_centers = out + (size_t)B_ * N_ * COUT;          // [B,K,3]
  float* out_nn      = out_centers + (size_t)B_ * K_ * 3;     // [B,N]

  hipLaunchKernelGGL(fps_kernel, dim3(B_), dim3(1024), 0, stream,
                     xyz, out_centers);
  hipLaunchKernelGGL(knn_group_kernel, dim3(B_ * (N_ / 128)), dim3(256), 0, stream,
                     xyz, features, out_centers, out_gf, out_nn);
}